// GIN_adj_79800492359958
// MI455X (gfx1250) — compile-verified
//
#include <hip/hip_runtime.h>

typedef __attribute__((ext_vector_type(2))) float v2f;
typedef __attribute__((ext_vector_type(8))) float v8f;
typedef __attribute__((ext_vector_type(4))) unsigned int u32x4;
typedef __attribute__((ext_vector_type(8))) int i32x8;
typedef __attribute__((ext_vector_type(4))) int i32x4;

#define BB   64
#define MM   512
#define HH   512
#define TILE 64
#define KT   32
#define LSTR 34   // LDS row stride in floats: 32 data + 2 pad (bank-conflict free)

// -----------------------------------------------------------------------------
// TDM: load a 2D f32 tile (dim0 = KT contiguous elems, dim1 = TILE rows,
// tensor row stride = HH elems) from global into LDS with +2-dword padding
// every 32 dwords (-> LDS row stride LSTR). Issued once per calling wave.
// -----------------------------------------------------------------------------
__device__ __forceinline__ void tdm_load_tile_2d(unsigned lds_byte_addr,
                                                 const float* gsrc) {
    unsigned long long ga = (unsigned long long)(const void*)gsrc;
    u32x4 g0;
    g0[0] = 1u;                                    // count=1 (valid user D#)
    g0[1] = lds_byte_addr;                         // lds_addr
    g0[2] = (unsigned)(ga & 0xFFFFFFFFu);          // global_addr[31:0]
    g0[3] = (unsigned)((ga >> 32) & 0x01FFFFFFu)   // global_addr[56:32]
            | (2u << 30);                          // type = 2 ("image")
    i32x8 g1;
    g1[0] = (int)((2u << 16)      // data_size = 4 bytes
                | (1u << 20)      // pad_enable
                | (4u << 22)      // pad_interval: every 32 dwords
                | (1u << 25));    // pad_amount: 2 dwords
    g1[1] = (int)(((unsigned)HH & 0xFFFFu) << 16);    // tensor_dim0 lo16
    g1[2] = (int)((((unsigned)HH >> 16) & 0xFFFFu)    // tensor_dim0 hi16
                | (((unsigned)MM & 0xFFFFu) << 16));  // tensor_dim1 lo16
    g1[3] = (int)((((unsigned)MM >> 16) & 0xFFFFu)    // tensor_dim1 hi16
                | ((unsigned)KT << 16));              // tile_dim0 = 32
    g1[4] = (int)((unsigned)TILE);                    // tile_dim1 = 64, tile_dim2 = 0
    g1[5] = (int)((unsigned)HH);                      // tensor_dim0_stride lo32
    g1[6] = 0;                                        // stride hi bits = 0
    g1[7] = 0;
    i32x4 z4 = {0, 0, 0, 0};
#if __clang_major__ >= 23
    i32x8 z8 = {0, 0, 0, 0, 0, 0, 0, 0};
    __builtin_amdgcn_tensor_load_to_lds(g0, g1, z4, z4, z8, 0);
#else
    __builtin_amdgcn_tensor_load_to_lds(g0, g1, z4, z4, 0);
#endif
}

// -----------------------------------------------------------------------------
// Kernel 1: Y[b] = (1+eps)*X[b] + A[b]^T @ X[b]
// Both LDS tiles K-contiguous ([row][k], stride LSTR) so every WMMA fragment
// is a single aligned ds_load_b64. Tiles are transposed on LDS write (global
// reads stay coalesced). Software-pipelined: global loads for tile kt+1 are
// issued before the WMMA phase of tile kt (register staging + double-buffered
// LDS), hiding HBM latency under compute.
// -----------------------------------------------------------------------------
__global__ __launch_bounds__(128)
void gin_msg_kernel(const float* __restrict__ A, const float* __restrict__ X,
                    const float* __restrict__ eps, float* __restrict__ Y) {
    __shared__ __align__(16) float As[2][TILE * LSTR];  // As[j][k] = A[k][tileJ+j]
    __shared__ __align__(16) float Bs[2][TILE * LSTR];  // Bs[h][k] = X[k][tileH+h]

    const int b     = blockIdx.z;
    const int tileJ = blockIdx.y * TILE;
    const int tileH = blockIdx.x * TILE;
    const int tid   = threadIdx.x;
    const int lane  = tid & 31;
    const int wave  = tid >> 5;
    const int waveM = (wave >> 1) * 32;
    const int waveN = (wave & 1) * 32;
    const int r     = lane & 15;
    const int kh    = lane >> 4;

    const float* Ab = A + (size_t)b * MM * MM;
    const float* Xb = X + (size_t)b * MM * HH;

    v8f acc[2][2] = {};
    float4 ra[4], rb[4];   // register-staged tile (A-part, X-part)

    auto issue_loads = [&](int k0) {
        #pragma unroll
        for (int i = 0; i < 4; ++i) {
            int f  = tid + i * 128;          // 512 float4 slots
            int kk = f >> 4;                 // 0..31
            int jv = (f & 15) << 2;          // 0..60
            ra[i] = *(const float4*)(Ab + (size_t)(k0 + kk) * MM + tileJ + jv);
            rb[i] = *(const float4*)(Xb + (size_t)(k0 + kk) * HH + tileH + jv);
        }
    };
    auto store_tile = [&](int buf) {
        #pragma unroll
        for (int i = 0; i < 4; ++i) {
            int f  = tid + i * 128;
            int kk = f >> 4;
            int jv = (f & 15) << 2;
            As[buf][(jv + 0) * LSTR + kk] = ra[i].x;
            As[buf][(jv + 1) * LSTR + kk] = ra[i].y;
            As[buf][(jv + 2) * LSTR + kk] = ra[i].z;
            As[buf][(jv + 3) * LSTR + kk] = ra[i].w;
            Bs[buf][(jv + 0) * LSTR + kk] = rb[i].x;
            Bs[buf][(jv + 1) * LSTR + kk] = rb[i].y;
            Bs[buf][(jv + 2) * LSTR + kk] = rb[i].z;
            Bs[buf][(jv + 3) * LSTR + kk] = rb[i].w;
        }
    };

    const int NK = MM / KT;                  // 16 k-tiles
    issue_loads(0);

    for (int kt = 0; kt < NK; ++kt) {
        const int cur = kt & 1;
        store_tile(cur);                     // waits on in-flight loads
        __syncthreads();                     // staged tile visible to all waves
        if (kt + 1 < NK) issue_loads((kt + 1) * KT);   // in flight during WMMAs

        const float* Ac = &As[cur][0];
        const float* Bc = &Bs[cur][0];
        #pragma unroll
        for (int kk = 0; kk < KT; kk += 4) {
            v2f af[2], bf[2];
            #pragma unroll
            for (int s = 0; s < 2; ++s)
                af[s] = *(const v2f*)&Ac[(waveM + s * 16 + r) * LSTR + kk + 2 * kh];
            #pragma unroll
            for (int t = 0; t < 2; ++t)
                bf[t] = *(const v2f*)&Bc[(waveN + t * 16 + r) * LSTR + kk + 2 * kh];
            #pragma unroll
            for (int s = 0; s < 2; ++s)
                #pragma unroll
                for (int t = 0; t < 2; ++t)
                    acc[s][t] = __builtin_amdgcn_wmma_f32_16x16x4_f32(
                        false, af[s], false, bf[t], (short)0, acc[s][t],
                        false, false);
        }
        __syncthreads();                     // all reads done before buf reuse
    }

    const float e1 = 1.0f + eps[0];
    float* Yb = Y + (size_t)b * MM * HH;
    #pragma unroll
    for (int s = 0; s < 2; ++s)
        #pragma unroll
        for (int t = 0; t < 2; ++t)
            #pragma unroll
            for (int rr = 0; rr < 8; ++rr) {
                int row = tileJ + waveM + s * 16 + rr + 8 * kh;
                int col = tileH + waveN + t * 16 + r;
                size_t idx = (size_t)row * HH + col;
                Yb[idx] = e1 * Xb[idx] + acc[s][t][rr];
            }
}

// -----------------------------------------------------------------------------
// Kernel 2/3: Out[b] = relu(In[b] @ W^T + bias)  [ * wmul[b][row] if wmul ]
// Both tiles are K-contiguous in global memory (In rows; W stored [out][in])
// so they are straight 2D TDM tiles: wave 0 issues double-buffered
// tensor_load_to_lds with LDS padding baked into the descriptor; compute
// overlaps the in-flight DMA via s_wait_tensorcnt.
// -----------------------------------------------------------------------------
__global__ __launch_bounds__(128)
void gin_fc_kernel(const float* __restrict__ In, const float* __restrict__ W,
                   const float* __restrict__ bias,
                   const float* __restrict__ wmul, float* __restrict__ Out) {
    __shared__ __align__(16) float As[2][TILE * LSTR];  // As[m][k] = In[tileM+m][k]
    __shared__ __align__(16) float Bs[2][TILE * LSTR];  // Bs[n][k] = W[tileN+n][k]

    const int b     = blockIdx.z;
    const int tileM = blockIdx.y * TILE;
    const int tileN = blockIdx.x * TILE;
    const int tid   = threadIdx.x;
    const int lane  = tid & 31;
    const int wave  = tid >> 5;
    const int waveM = (wave >> 1) * 32;
    const int waveN = (wave & 1) * 32;
    const int r     = lane & 15;
    const int kh    = lane >> 4;

    const float* Ib = In + (size_t)b * MM * HH;

    v8f acc[2][2] = {};
    const int NK = HH / KT;  // 16 k-tiles

    if (wave == 0) {
        tdm_load_tile_2d((unsigned)(size_t)&As[0][0], Ib + (size_t)tileM * HH);
        tdm_load_tile_2d((unsigned)(size_t)&Bs[0][0], W + (size_t)tileN * HH);
    }

    for (int kt = 0; kt < NK; ++kt) {
        const int cur = kt & 1;
        if (wave == 0) {
            if (kt + 1 < NK) {
                const int k0n = (kt + 1) * KT;
                tdm_load_tile_2d((unsigned)(size_t)&As[cur ^ 1][0],
                                 Ib + (size_t)tileM * HH + k0n);
                tdm_load_tile_2d((unsigned)(size_t)&Bs[cur ^ 1][0],
                                 W + (size_t)tileN * HH + k0n);
                __builtin_amdgcn_s_wait_tensorcnt(2);  // current tile landed
            } else {
                __builtin_amdgcn_s_wait_tensorcnt(0);  // last tile landed
            }
        }
        __syncthreads();

        const float* Ac = &As[cur][0];
        const float* Bc = &Bs[cur][0];
        #pragma unroll
        for (int kk = 0; kk < KT; kk += 4) {
            v2f af[2], bf[2];
            #pragma unroll
            for (int s = 0; s < 2; ++s)
                af[s] = *(const v2f*)&Ac[(waveM + s * 16 + r) * LSTR + kk + 2 * kh];
            #pragma unroll
            for (int t = 0; t < 2; ++t)
                bf[t] = *(const v2f*)&Bc[(waveN + t * 16 + r) * LSTR + kk + 2 * kh];
            #pragma unroll
            for (int s = 0; s < 2; ++s)
                #pragma unroll
                for (int t = 0; t < 2; ++t)
                    acc[s][t] = __builtin_amdgcn_wmma_f32_16x16x4_f32(
                        false, af[s], false, bf[t], (short)0, acc[s][t],
                        false, false);
        }
        __syncthreads();   // all waves done reading before buffer is re-filled
    }

    float* Ob = Out + (size_t)b * MM * HH;
    #pragma unroll
    for (int s = 0; s < 2; ++s)
        #pragma unroll
        for (int t = 0; t < 2; ++t) {
            int colBase = tileN + waveN + t * 16 + r;
            float bv = bias[colBase];
            #pragma unroll
            for (int rr = 0; rr < 8; ++rr) {
                int row = tileM + waveM + s * 16 + rr + 8 * kh;
                float v = acc[s][t][rr] + bv;
                v = v > 0.0f ? v : 0.0f;
                if (wmul) v *= wmul[(size_t)b * MM + row];
                Ob[(size_t)row * HH + colBase] = v;
            }
        }
}

// -----------------------------------------------------------------------------
// Launch: inputs are (X, E, A, w, eps, W0_w, W0_b, W1_w, W1_b)
// -----------------------------------------------------------------------------
extern "C" void kernel_launch(void* const* d_in, const int* in_sizes, int n_in,
                              void* d_out, int out_size, void* d_ws,
                              size_t ws_size, hipStream_t stream) {
    const float* X   = (const float*)d_in[0];
    const float* A   = (const float*)d_in[2];
    const float* w   = (const float*)d_in[3];
    const float* eps = (const float*)d_in[4];
    const float* W0w = (const float*)d_in[5];
    const float* W0b = (const float*)d_in[6];
    const float* W1w = (const float*)d_in[7];
    const float* W1b = (const float*)d_in[8];
    float* out = (float*)d_out;

    float* Y = (float*)d_ws;                       // 64 MB
    float* Z = Y + (size_t)BB * MM * HH;           // next 64 MB

    dim3 grid(HH / TILE, MM / TILE, BB);           // (8, 8, 64)
    dim3 block(128);

    gin_msg_kernel<<<grid, block, 0, stream>>>(A, X, eps, Y);
    gin_fc_kernel<<<grid, block, 0, stream>>>(Y, W0w, W0b, nullptr, Z);
    gin_fc_kernel<<<grid, block, 0, stream>>>(Z, W1w, W1b, w, out);
}